// HungarianMatcher_1735166787973
// MI455X (gfx1250) — compile-verified
//
#include <hip/hip_runtime.h>
#include <hip/hip_bf16.h>
#include <math.h>

#define NGT 256   // gt rows
#define NPR 1024  // pred cols
#define NL  91    // logits dim

typedef __attribute__((ext_vector_type(2))) float v2f;
typedef __attribute__((ext_vector_type(8))) float v8f;
typedef __attribute__((ext_vector_type(4))) unsigned int v4u;
typedef __attribute__((ext_vector_type(8))) int v8i;
typedef __attribute__((ext_vector_type(4))) int v4i;

#if defined(__HIP_DEVICE_COMPILE__) && __has_builtin(__builtin_amdgcn_tensor_load_to_lds)
#define USE_TDM 1
#else
#define USE_TDM 0
#endif

#if USE_TDM
// Issue a 2-D TDM tile load: rows x NL f32 elements, row-major contiguous
// (stride NL), into LDS at lds_byte_off. D# layout per CDNA5 ISA.
// This toolchain exposes the 6-arg builtin: (g0 v4u, g1 v8i, g2 v4i, g3 v4i,
// extra v8i, cpol i32); groups 2/3 disabled (2-D tensor), extra passed as 0.
__device__ __forceinline__ void tdm_load_tile_2d(const float* gsrc, int rows,
                                                 unsigned int lds_byte_off) {
  unsigned long long ga = (unsigned long long)(const void*)gsrc;
  v4u g0;
  g0[0] = 1u;                                                      // count=1 (valid), user mode
  g0[1] = lds_byte_off;                                            // lds_addr
  g0[2] = (unsigned int)(ga & 0xFFFFFFFFull);                      // global_addr[31:0]
  g0[3] = (unsigned int)((ga >> 32) & 0x1FFFFFFull) | (2u << 30);  // global_addr[56:32] | type=2
  v8i g1;
  g1[0] = (int)(2u << 16);               // data_size=4B, no multicast, no pad/iterate
  g1[1] = (int)((unsigned)NL << 16);     // tensor_dim0[15:0] @ bit48
  g1[2] = (int)((unsigned)rows << 16);   // tensor_dim0 hi | tensor_dim1[15:0]
  g1[3] = (int)((unsigned)NL << 16);     // tensor_dim1 hi | tile_dim0=NL
  g1[4] = rows;                          // tile_dim1=rows, tile_dim2=0
  g1[5] = NL;                            // tensor_dim0_stride[31:0]
  g1[6] = 0;                             // stride hi | dim1_stride lo (unused, 2D)
  g1[7] = 0;
  v4i g2 = {0, 0, 0, 0};
  v4i g3 = {0, 0, 0, 0};
  v8i g4 = {0, 0, 0, 0, 0, 0, 0, 0};
  __builtin_amdgcn_tensor_load_to_lds(g0, g1, g2, g3, g4, 0);
}
#endif

// One workgroup = 256 threads = 8 waves; each wave owns a 16x16 tile of the
// 256x1024 cost matrix. WG covers 16 rows x 128 cols. Squared Euclidean
// distance via V_WMMA_F32_16X16X4_F32 (A=[gx,gy,|g|^2,1], B=[-2px,-2py,1,|p|^2]),
// L1 logits cost via VALU from LDS-staged tiles.
__global__ __launch_bounds__(256) void hm_cost_tiles_kernel(
    const float* __restrict__ cgt, const float* __restrict__ cpr,
    const float* __restrict__ lgt, const float* __restrict__ lpr,
    float* __restrict__ costm, float* __restrict__ distm) {
#if defined(__HIP_DEVICE_COMPILE__)
  __shared__ float lgs[16 * NL];    // gt logits tile
  __shared__ float lps[128 * NL];   // pred logits tile
  const int t = threadIdx.x;
  const int rowBase = blockIdx.y * 16;
  const int colBase = blockIdx.x * 128;

#if USE_TDM
  if (t < 32) {  // wave 0 issues both DMA descriptors, waits, then WG barrier
    tdm_load_tile_2d(lgt + rowBase * NL, 16,
                     (unsigned int)(unsigned long long)(void*)lgs);
    tdm_load_tile_2d(lpr + colBase * NL, 128,
                     (unsigned int)(unsigned long long)(void*)lps);
    __builtin_amdgcn_s_wait_tensorcnt(0);
  }
#else
  for (int idx = t; idx < 16 * NL; idx += 256) lgs[idx] = lgt[rowBase * NL + idx];
  for (int idx = t; idx < 128 * NL; idx += 256) lps[idx] = lpr[colBase * NL + idx];
#endif
  __syncthreads();

  const int w = t >> 5;        // wave id 0..7 -> 16-col subtile
  const int lane = t & 31;
  const int half = lane >> 4;  // 0: K=0,1 slots / M=0..7 of D; 1: K=2,3 / M=8..15
  const int l16 = lane & 15;
  const int myN = colBase + w * 16 + l16;

  // A 16x4 layout: lanes0-15 VGPR0/1 = K0/K1, lanes16-31 = K2/K3 (M = lane%16)
  // B 4x16 layout: lanes0-15 VGPR0/1 = K0/K1 rows (N = lane%16), lanes16-31 = K2/K3
  float gx = cgt[(rowBase + l16) * 2 + 0];
  float gy = cgt[(rowBase + l16) * 2 + 1];
  float px = cpr[myN * 2 + 0];
  float py = cpr[myN * 2 + 1];
  v2f A, B;
  if (half == 0) {
    A[0] = gx; A[1] = gy;
    B[0] = -2.0f * px; B[1] = -2.0f * py;
  } else {
    A[0] = gx * gx + gy * gy; A[1] = 1.0f;
    B[0] = 1.0f; B[1] = px * px + py * py;
  }
  v8f d2 = {};
  d2 = __builtin_amdgcn_wmma_f32_16x16x4_f32(false, A, false, B, (short)0, d2,
                                             false, false);

  // L1 over 91 logit dims; D layout: VGPR r -> M = r + 8*half, N = lane%16
  const int mOff = half * 8;
  const float* pr = &lps[(w * 16 + l16) * NL];
  const float* gr = &lgs[mOff * NL];
  float l1[8] = {0.f, 0.f, 0.f, 0.f, 0.f, 0.f, 0.f, 0.f};
  for (int k = 0; k < NL; ++k) {
    float pv = pr[k];
#pragma unroll
    for (int r = 0; r < 8; ++r) l1[r] += fabsf(gr[r * NL + k] - pv);
  }

#pragma unroll
  for (int r = 0; r < 8; ++r) {
    int mg = rowBase + mOff + r;
    float d = sqrtf(fmaxf(d2[r], 0.0f));
    distm[mg * NPR + myN] = d;
    costm[mg * NPR + myN] = d + l1[r];
  }
#endif  // __HIP_DEVICE_COMPILE__
}

// Jonker-Volgenant LSA, one workgroup of 1024 threads (32 wave32s), thread t
// owns column j=t+1. v[j]/minv[j] live in registers (only owner touches them);
// u/p/way/used in LDS. Argmin via wave shuffles + cross-wave LDS reduce,
// tie-break to smallest j (matches np.argmin over ascending free list).
__global__ __launch_bounds__(1024) void hm_lsa_kernel(
    const float* __restrict__ costm, const float* __restrict__ distm,
    float* __restrict__ out) {
  const int n = NGT, m = NPR;
  const double INFD = 1e18;
  __shared__ double u_[NGT + 1];
  __shared__ int p_[NPR + 1];
  __shared__ int way_[NPR + 1];
  __shared__ unsigned char used_[NPR + 1];
  __shared__ double red_v[32];
  __shared__ int red_j[32];
  __shared__ int sh_j0;
  __shared__ double sh_delta;
  __shared__ int col_[NGT];
  __shared__ double csum_[32];

  const int t = threadIdx.x;
  const int j = t + 1;
  const int lane = t & 31;
  const int wid = t >> 5;

  if (t <= NGT) u_[t] = 0.0;
  p_[j] = 0;
  way_[j] = 0;
  if (t == 0) { p_[0] = 0; way_[0] = 0; }
  double vj = 0.0;  // v[j], register-resident
  __syncthreads();

  for (int i = 1; i <= n; ++i) {
    double mv = INFD;  // minv[j]
    int wayj = 0;
    used_[j] = 0;
    if (t == 0) { p_[0] = i; used_[0] = 0; sh_j0 = 0; }
    __syncthreads();
    for (;;) {
      int j0 = sh_j0;
      if (t == 0) used_[j0] = 1;
      __syncthreads();
      int i0 = p_[j0];
      bool freecol = (used_[j] == 0);
      double myv = INFD;
      if (freecol) {
        double cur = (double)costm[(i0 - 1) * m + (j - 1)] - u_[i0] - vj;
        if (cur < mv) { mv = cur; wayj = j0; }
        myv = mv;
      }
      int myj = j;
      for (int off = 16; off > 0; off >>= 1) {
        double ov = __shfl_down(myv, off, 32);
        int oj = __shfl_down(myj, off, 32);
        if (ov < myv || (ov == myv && oj < myj)) { myv = ov; myj = oj; }
      }
      if (lane == 0) { red_v[wid] = myv; red_j[wid] = myj; }
      __syncthreads();
      if (t < 32) {
        double rv = red_v[t];
        int rj = red_j[t];
        for (int off = 16; off > 0; off >>= 1) {
          double ov = __shfl_down(rv, off, 32);
          int oj = __shfl_down(rj, off, 32);
          if (ov < rv || (ov == rv && oj < rj)) { rv = ov; rj = oj; }
        }
        if (t == 0) { sh_delta = rv; sh_j0 = rj; }
      }
      __syncthreads();
      double delta = sh_delta;
      int j1 = sh_j0;
      if (freecol) {
        mv -= delta;            // minv[~used] -= delta
      } else {
        u_[p_[j]] += delta;     // distinct p over used cols -> no write race
        vj -= delta;            // v[used] -= delta
      }
      if (t == 0) u_[p_[0]] += delta;  // virtual column 0 is always used
      way_[j] = wayj;
      __syncthreads();  // updates + way_ visible before next scan / augment
      if (p_[j1] == 0) break;  // uniform: shared state
    }
    if (t == 0) {  // augment alternating path
      int jj = sh_j0;
      while (jj != 0) {
        int jp = way_[jj];
        p_[jj] = p_[jp];
        jj = jp;
      }
    }
    __syncthreads();
  }

  // Extract col_ind, emit outputs, and reduce cost = sum dist[i, col[i]]
  if (p_[j] > 0) col_[p_[j] - 1] = j - 1;
  __syncthreads();
  if (t < NGT) {
    out[t] = (float)t;                // row_ind = arange(n)
    out[NGT + t] = (float)col_[t];    // col_ind
  }
  double c = (t < NGT) ? (double)distm[t * m + col_[t]] : 0.0;
  for (int off = 16; off > 0; off >>= 1) c += __shfl_down(c, off, 32);
  if (lane == 0) csum_[wid] = c;
  __syncthreads();
  if (t == 0) {
    double s = 0.0;
    for (int wv = 0; wv < 32; ++wv) s += csum_[wv];
    out[2 * NGT] = (float)s;
  }
}

extern "C" void kernel_launch(void* const* d_in, const int* in_sizes, int n_in,
                              void* d_out, int out_size, void* d_ws,
                              size_t ws_size, hipStream_t stream) {
  const float* cgt = (const float*)d_in[0];  // (256,2)
  const float* cpr = (const float*)d_in[1];  // (1024,2)
  const float* lgt = (const float*)d_in[2];  // (256,91)
  const float* lpr = (const float*)d_in[3];  // (1024,91)
  float* costm = (float*)d_ws;               // 256*1024 f32
  float* distm = costm + NGT * NPR;          // 256*1024 f32

  hm_cost_tiles_kernel<<<dim3(NPR / 128, NGT / 16), 256, 0, stream>>>(
      cgt, cpr, lgt, lpr, costm, distm);
  hm_lsa_kernel<<<1, 1024, 0, stream>>>(costm, distm, (float*)d_out);
}